// GraySmoothnessRewardModel_42271068127739
// MI455X (gfx1250) — compile-verified
//
#include <hip/hip_runtime.h>

typedef __attribute__((ext_vector_type(2))) float v2f;
typedef __attribute__((ext_vector_type(8))) float v8f;

#define IMG_W 1024
#define IMG_H 1024
#define RH    64            // owned grad_mag rows per block
#define NROWB (RH + 4)      // image rows streamed per block
#define ROWPAD 1028         // 1024 + 4 pad floats; 4112 B per row, 16B aligned

// -----------------------------------------------------------------------------
// Main fused kernel: sobel -> grad_mag -> |diff_x|,|diff_y| partial sums.
// grid = (ceil(1022/RH)=16, 64 images), block = 256 threads, thread owns 4 cols.
// Per-image sums accumulated as 2^32 fixed-point u64 atomics (deterministic).
// -----------------------------------------------------------------------------
__global__ __launch_bounds__(256) void reward_main(
    const float* __restrict__ img, unsigned long long* __restrict__ acc)
{
  __shared__ __align__(16) float ibuf[2][ROWPAD];  // double-buffered image rows
  __shared__ __align__(16) float gbuf[ROWPAD];     // current grad_mag row
  __shared__ float red[512];                       // block reduction staging

  const int t  = threadIdx.x;
  const int x4 = t << 2;                 // first owned column
  const int y0 = blockIdx.x * RH;        // first owned grad_mag row
  const int im = blockIdx.y;
  const float* base = img + (size_t)im * (IMG_W * IMG_H);

  // column ownership masks (hoisted; only thread 255 is partially masked)
  const bool cx0 = (x4 + 0 <= 1019), cx1 = (x4 + 1 <= 1019);
  const bool cx2 = (x4 + 2 <= 1019), cx3 = (x4 + 3 <= 1019);
  const bool cy0 = (x4 + 0 <= 1021), cy1 = (x4 + 1 <= 1021);
  const bool cy2 = (x4 + 2 <= 1021), cy3 = (x4 + 3 <= 1021);

  float dA[4] = {0,0,0,0}, dB[4] = {0,0,0,0};   // horizontal-diff history (rows r, r+1)
  float sA[4] = {0,0,0,0}, sB[4] = {0,0,0,0};   // horizontal-smooth history
  float gm1[4] = {0,0,0,0}, gm2[4] = {0,0,0,0}; // grad_mag history (rows r-1, r-2)
  float accx = 0.0f, accy = 0.0f;

  // Preload image row 0 (async DMA global -> LDS, b128 per lane = 4 floats)
  {
    int yr = y0; if (yr > IMG_H - 1) yr = IMG_H - 1;
    const float* gp = base + (size_t)yr * IMG_W + x4;
    unsigned int lp = (unsigned int)(size_t)&ibuf[0][x4];
    asm volatile("global_load_async_to_lds_b128 %0, %1, off"
                 :: "v"(lp), "v"(gp) : "memory");
  }

  for (int k = 0; k < NROWB; ++k) {
    // Prefetch next row into the other buffer, then wait for row k's DMA.
    if (k + 1 < NROWB) {
      int yr = y0 + k + 1; if (yr > IMG_H - 1) yr = IMG_H - 1;
      const float* gp = base + (size_t)yr * IMG_W + x4;
      unsigned int lp = (unsigned int)(size_t)&ibuf[(k + 1) & 1][x4];
      asm volatile("global_load_async_to_lds_b128 %0, %1, off"
                   :: "v"(lp), "v"(gp) : "memory");
      asm volatile("s_wait_asynccnt 0x1" ::: "memory");
    } else {
      asm volatile("s_wait_asynccnt 0x0" ::: "memory");
    }
    __syncthreads();   // row k fully resident in LDS for all waves

    const float* rb = ibuf[k & 1];
    float i0 = rb[x4 + 0], i1 = rb[x4 + 1], i2 = rb[x4 + 2], i3 = rb[x4 + 3];
    float i4 = rb[x4 + 4], i5 = rb[x4 + 5];   // neighbor overlap (padded)

    // Separable sobel row terms (unscaled; /8 and /2 folded into final divide)
    float d[4], s[4];
    d[0] = i2 - i0; d[1] = i3 - i1; d[2] = i4 - i2; d[3] = i5 - i3;
    s[0] = (i0 + i2) + 2.0f * i1; s[1] = (i1 + i3) + 2.0f * i2;
    s[2] = (i2 + i4) + 2.0f * i3; s[3] = (i3 + i5) + 2.0f * i4;

    float cur[4] = {0,0,0,0};
    if (k >= 2) {       // grad_mag row r = k-2 (uses image rows r..r+2)
#pragma unroll
      for (int j = 0; j < 4; ++j) {
        float gx = (dA[j] + d[j]) + 2.0f * dB[j];  // vertical [1,2,1] of d
        float gy = s[j] - sA[j];                   // vertical [-1,0,1] of s
        cur[j] = sqrtf(gx * gx + gy * gy);
      }
      gbuf[x4 + 0] = cur[0]; gbuf[x4 + 1] = cur[1];
      gbuf[x4 + 2] = cur[2]; gbuf[x4 + 3] = cur[3];
    }
    __syncthreads();   // grad_mag row visible for neighbor reads

    if (k >= 2) {
      const int rl = k - 2, rg = y0 + rl;
      if (rl < RH && rg <= 1021) {               // |gdx| terms for row rg
        float n0 = gbuf[x4 + 4], n1 = gbuf[x4 + 5];
        float a0 = cx0 ? fabsf(cur[2] - cur[0]) : 0.0f;
        float a1 = cx1 ? fabsf(cur[3] - cur[1]) : 0.0f;
        float a2 = cx2 ? fabsf(n0 - cur[2]) : 0.0f;
        float a3 = cx3 ? fabsf(n1 - cur[3]) : 0.0f;
        accx += (a0 + a1) + (a2 + a3);
      }
      if (k >= 4) {
        const int il = k - 4, ig = y0 + il;
        if (il < RH && ig <= 1019) {             // |gdy| terms for row ig
          float b0 = cy0 ? fabsf(cur[0] - gm2[0]) : 0.0f;
          float b1 = cy1 ? fabsf(cur[1] - gm2[1]) : 0.0f;
          float b2 = cy2 ? fabsf(cur[2] - gm2[2]) : 0.0f;
          float b3 = cy3 ? fabsf(cur[3] - gm2[3]) : 0.0f;
          accy += (b0 + b1) + (b2 + b3);
        }
      }
#pragma unroll
      for (int j = 0; j < 4; ++j) { gm2[j] = gm1[j]; gm1[j] = cur[j]; }
    }
#pragma unroll
    for (int j = 0; j < 4; ++j) { dA[j] = dB[j]; dB[j] = d[j];
                                  sA[j] = sB[j]; sB[j] = s[j]; }
  }

  // ---------------- block reduction via V_WMMA_F32_16X16X4_F32 ----------------
  red[t] = accx;
  red[256 + t] = accy;
  __syncthreads();

  if (t < 32) {                       // whole wave 0, EXEC all ones
    const int half = t >> 4;          // 0 -> K=0,1 ; 1 -> K=2,3
    const int m    = t & 15;          // A-matrix row
    v8f cx = {0,0,0,0,0,0,0,0};
    v8f cy = {0,0,0,0,0,0,0,0};
    const v2f bones = {1.0f, 1.0f};   // B = all ones => D[m][n] = row-sum of A
#pragma unroll
    for (int chunk = 0; chunk < 4; ++chunk) {
      int idx = chunk * 64 + m * 4 + half * 2;      // A[m][k] = red[chunk*64+m*4+k]
      v2f ax = { red[idx],       red[idx + 1] };
      v2f ay = { red[256 + idx], red[256 + idx + 1] };
      cx = __builtin_amdgcn_wmma_f32_16x16x4_f32(false, ax, false, bones,
                                                 (short)0, cx, false, false);
      cy = __builtin_amdgcn_wmma_f32_16x16x4_f32(false, ay, false, bones,
                                                 (short)0, cy, false, false);
    }
    // lane 0 holds D[0..7][0], lane 16 holds D[8..15][0]
    float sx = ((cx[0] + cx[1]) + (cx[2] + cx[3])) + ((cx[4] + cx[5]) + (cx[6] + cx[7]));
    float sy = ((cy[0] + cy[1]) + (cy[2] + cy[3])) + ((cy[4] + cy[5]) + (cy[6] + cy[7]));
    sx += __shfl_down(sx, 16);
    sy += __shfl_down(sy, 16);
    if (t == 0) {
      unsigned long long qx = (unsigned long long)((double)sx * 4294967296.0);
      unsigned long long qy = (unsigned long long)((double)sy * 4294967296.0);
      atomicAdd(&acc[(size_t)im * 2 + 0], qx);   // deterministic integer atomics
      atomicAdd(&acc[(size_t)im * 2 + 1], qy);
    }
  }
}

// -----------------------------------------------------------------------------
// Finalize: mx = sumx'/(16*N), my = sumy'/(16*N), out = 1/(1+mx+my)
// (factor 16 folds the sobel /8 and diff /2 scales; N = 1022*1020 terms each)
// -----------------------------------------------------------------------------
__global__ void reward_finalize(const unsigned long long* __restrict__ acc,
                                float* __restrict__ out)
{
  int i = threadIdx.x;
  if (i < 64) {
    const double invScale = 1.0 / 4294967296.0;
    double sx = (double)acc[2 * i + 0] * invScale;
    double sy = (double)acc[2 * i + 1] * invScale;
    const double denom = 16.0 * 1022.0 * 1020.0;
    double mx = sx / denom;
    double my = sy / denom;
    out[i] = (float)(1.0 / (1.0 + mx + my));
  }
}

extern "C" void kernel_launch(void* const* d_in, const int* in_sizes, int n_in,
                              void* d_out, int out_size, void* d_ws, size_t ws_size,
                              hipStream_t stream)
{
  (void)in_sizes; (void)n_in; (void)out_size; (void)ws_size;
  const float* img = (const float*)d_in[0];
  float* out = (float*)d_out;
  unsigned long long* acc = (unsigned long long*)d_ws;

  hipMemsetAsync(acc, 0, 64 * 2 * sizeof(unsigned long long), stream);

  dim3 grid((1022 + RH - 1) / RH, 64);   // (16 row-strips, 64 images)
  reward_main<<<grid, 256, 0, stream>>>(img, acc);
  reward_finalize<<<1, 64, 0, stream>>>(acc, out);
}